// reduction_layer_8813272891666
// MI455X (gfx1250) — compile-verified
//
#include <hip/hip_runtime.h>
#include <stdint.h>

#define BATCH 2
#define NPTS  8192
#define EDIM  256
#define NHN   32
#define NKEEP 4096

typedef __attribute__((ext_vector_type(2))) float v2f;
typedef __attribute__((ext_vector_type(8))) float v8f;
typedef __attribute__((ext_vector_type(4))) unsigned int v4u;
typedef __attribute__((ext_vector_type(8))) int v8i;
typedef __attribute__((ext_vector_type(4))) int v4i;

// ---------------- ws layout (float units) ----------------
// [0, 8192)      partial column sums  [b][16][256]
// [8192, 8704)   invgm[b][256]  = 1/|mean_n x|
// [8704, 16896)  keep indices   [b][4096] (int)

// Stage 1: partial column sums of x over n (no atomics -> deterministic)
__global__ void colsum_part_kernel(const float* __restrict__ x,
                                   float* __restrict__ part) {
  const int b = blockIdx.y, p = blockIdx.x, e = threadIdx.x; // 16 parts x 512 rows
  const float* xp = x + (((size_t)b * NPTS + (size_t)p * 512) << 8) + e;
  float acc = 0.f;
#pragma unroll 8
  for (int r = 0; r < 512; ++r) acc += xp[(size_t)r << 8];
  part[(b * 16 + p) * 256 + e] = acc;
}

// Stage 2: invgm = 1/|sum/N|
__global__ void make_invgm_kernel(const float* __restrict__ part,
                                  float* __restrict__ invgm) {
  const int i = threadIdx.x;         // 0..511
  const int b = i >> 8, e = i & 255;
  float s = 0.f;
  for (int p = 0; p < 16; ++p) s += part[(b * 16 + p) * 256 + e];
  invgm[i] = 1.0f / fabsf(s * (1.0f / (float)NPTS));
}

// TDM gather-mode descriptor: DMA 16 rows (16-bit indices) of 256 f32 from
// global tensor [8192][256] into LDS at lds_off. Tracked by TENSORcnt.
__device__ __forceinline__ void tdm_gather16(const int* __restrict__ idx16,
                                             unsigned int lds_off,
                                             const float* __restrict__ gbase) {
  const unsigned long long ga = (unsigned long long)(uintptr_t)gbase;
  v4u g0;
  g0.x = 0x80000001u;                    // count=1, gather_mode=1, 16-bit idx
  g0.y = lds_off;                        // LDS byte address of tile start
  g0.z = (unsigned int)ga;               // global_addr[31:0]
  g0.w = (unsigned int)((ga >> 32) & 0x1FFFFFFu) | (2u << 30); // addr[56:32]|type=2
  v8i g1;
  g1[0] = (2 << 16);                     // wg_mask=0, data_size=4B
  g1[1] = (int)(((unsigned)EDIM & 0xFFFFu) << 16);  // tensor_dim0[15:0]
  g1[2] = (int)(((unsigned)NPTS & 0xFFFFu) << 16);  // td0 hi=0, tensor_dim1[15:0]
  g1[3] = (int)((unsigned)EDIM << 16);   // td1 hi=0, tile_dim0=256
  g1[4] = 16;                            // tile_dim1 = #valid indices
  g1[5] = EDIM;                          // tensor_dim0_stride low32
  g1[6] = 0;
  g1[7] = 0;
  v4i g2, g3;
#pragma unroll
  for (int w = 0; w < 4; ++w) {
    g2[w] = (idx16[2 * w] & 0xFFFF) | (idx16[2 * w + 1] << 16);
    g3[w] = (idx16[8 + 2 * w] & 0xFFFF) | (idx16[8 + 2 * w + 1] << 16);
  }
#if __clang_major__ >= 23
  v8i gz = {};
  __builtin_amdgcn_tensor_load_to_lds(g0, g1, g2, g3, gz, 0);
#else
  __builtin_amdgcn_tensor_load_to_lds(g0, g1, g2, g3, 0);
#endif
}

// Stage 3: fused KNN (WMMA f32 16x16x4 distance tiles) + TDM-gathered
// neighbor-std + local_dist.  Block = 128 threads (4 waves), 16 queries/wave.
__global__ __launch_bounds__(128) void knn_std_kernel(
    const float* __restrict__ x, const float* __restrict__ coords,
    const float* __restrict__ invgm, float* __restrict__ out_ld) {
  __shared__ float sqqs[4][16];        // per-wave query squared norms
  __shared__ float tile[4][16][16];    // per-wave 16x16 d2 tile
  __shared__ float listd[4][32][32];   // per-lane sorted top-32 distances
  __shared__ int   listi[4][32][32];   // per-lane sorted top-32 indices
  __shared__ int   nbr[4][16][32];     // merged neighbor indices per query
  extern __shared__ float dynbuf[];    // 4 waves x 8192 f32 TDM gather buffers

  const int wave = threadIdx.x >> 5;
  const int lane = threadIdx.x & 31;
  const int b = blockIdx.x >> 7;                 // 128 blocks per batch
  const int qbase = (blockIdx.x & 127) * 64 + wave * 16;
  const int hi = lane >> 4, lo = lane & 15;

  const float* cx = coords + (size_t)b * 2 * NPTS;     // coords[b][0][n]
  const float* cy = cx + NPTS;                          // coords[b][1][n]
  const float* xb = x + ((size_t)b * NPTS << 8);

  // ---- A matrix (16x4 f32): M=query (lanes 0-15), VGPR0:K0/K2, VGPR1:K1/K3
  float qx = 0.f, qy = 0.f;
  if (hi == 0) {
    qx = cx[qbase + lo];
    qy = cy[qbase + lo];
    sqqs[wave][lo] = qx * qx + qy * qy;
  }
  v2f amat;
  amat.x = -2.0f * qx;   // K=0 rows; lanes 16-31 carry K=2 == 0
  amat.y = -2.0f * qy;   // K=1 rows; lanes 16-31 carry K=3 == 0
  __syncthreads();

  float sqq_r[8];
#pragma unroll
  for (int r = 0; r < 8; ++r) sqq_r[r] = sqqs[wave][r + hi * 8];

  // per-lane top-32 state (2 lanes per query, 8 candidates per lane per tile)
  int cnt = 0;
  float worst = 3.402823e38f;
  const int q = lane >> 1, sub = lane & 1;
  float* myd = listd[wave][lane];
  int*   myi = listi[wave][lane];

  for (int chunk = 0; chunk < NPTS / 16; ++chunk) {
    const int n0 = chunk * 16;
    const float bx = cx[n0 + lo];
    const float by = cy[n0 + lo];
    const float sqc = bx * bx + by * by;
    v2f bmat;
    bmat.x = hi ? 0.f : bx;   // K=0 row on lanes 0-15, K=2 row zero
    bmat.y = hi ? 0.f : by;   // K=1 row / K=3 zero
    v8f cm;
#pragma unroll
    for (int r = 0; r < 8; ++r) cm[r] = sqq_r[r] + sqc;

    // d2 tile: D[M][N] = sq_q[M] + sq_c[N] - 2*(qx*bx + qy*by)
    v8f d = __builtin_amdgcn_wmma_f32_16x16x4_f32(
        false, amat, false, bmat, (short)0, cm, false, false);

#pragma unroll
    for (int r = 0; r < 8; ++r) tile[wave][r + hi * 8][lo] = d[r];
    asm volatile("" ::: "memory");   // same-wave DS is in-order; block reordering

    // maintain per-lane sorted top-32 (keep rolled: emit divergent code once)
#pragma clang loop unroll(disable)
    for (int t = 0; t < 8; ++t) {
      const int col = sub * 8 + t;
      const float cand = tile[wave][q][col];
      const int gidx = n0 + col;
      if (cnt < NHN) {
        int p = cnt++;
        while (p > 0 && myd[p - 1] > cand) {
          myd[p] = myd[p - 1]; myi[p] = myi[p - 1]; --p;
        }
        myd[p] = cand; myi[p] = gidx;
        if (cnt == NHN) worst = myd[NHN - 1];
      } else if (cand < worst) {
        int p = NHN - 1;
        while (p > 0 && myd[p - 1] > cand) {
          myd[p] = myd[p - 1]; myi[p] = myi[p - 1]; --p;
        }
        myd[p] = cand; myi[p] = gidx;
        worst = myd[NHN - 1];
      }
    }
    asm volatile("" ::: "memory");
  }

  // ---- merge the two per-lane lists of each query (lane sub==0 does it)
  if (sub == 0) {
    const float* d0 = listd[wave][lane];
    const int*   j0 = listi[wave][lane];
    const float* d1 = listd[wave][lane + 1];
    const int*   j1 = listi[wave][lane + 1];
    int a0 = 0, a1 = 0;
#pragma clang loop unroll(disable)
    for (int k = 0; k < NHN; ++k) {
      const float va = (a0 < NHN) ? d0[a0] : 3.402823e38f;
      const float vb = (a1 < NHN) ? d1[a1] : 3.402823e38f;
      const int   ia = (a0 < NHN) ? j0[a0] : 0x7fffffff;
      const int   ib = (a1 < NHN) ? j1[a1] : 0x7fffffff;
      const bool takeA = (va < vb) || (va == vb && ia < ib);
      nbr[wave][q][k] = takeA ? ia : ib;
      if (takeA) ++a0; else ++a1;
    }
  }
  asm volatile("" ::: "memory");

  // ---- neighbor std + local_dist; neighbors DMA'd by TDM gather descriptors.
  // Each lane owns 8 contiguous features (lane*8 .. lane*8+7) -> ds_load_b128.
  float invg[8];
#pragma unroll
  for (int j = 0; j < 8; ++j) invg[j] = invgm[b * 256 + lane * 8 + j];

  float* gbuf = dynbuf + (size_t)wave * 8192;   // two 4096-float halves

  for (int qq = 0; qq < 16; ++qq) {
    const int* nb = &nbr[wave][qq][0];
    tdm_gather16(nb,      (unsigned int)(size_t)(void*)gbuf,          xb);
    tdm_gather16(nb + 16, (unsigned int)(size_t)(void*)(gbuf + 4096), xb);

    float s[8], ss[8];
#pragma unroll
    for (int j = 0; j < 8; ++j) { s[j] = 0.f; ss[j] = 0.f; }

    __builtin_amdgcn_s_wait_tensorcnt((short)1);   // half 0 landed
    asm volatile("" ::: "memory");
#pragma unroll 4
    for (int k = 0; k < 16; ++k) {
      const float4 va = *(const float4*)(gbuf + k * EDIM + lane * 8);
      const float4 vb = *(const float4*)(gbuf + k * EDIM + lane * 8 + 4);
      s[0] += va.x; ss[0] += va.x * va.x;  s[1] += va.y; ss[1] += va.y * va.y;
      s[2] += va.z; ss[2] += va.z * va.z;  s[3] += va.w; ss[3] += va.w * va.w;
      s[4] += vb.x; ss[4] += vb.x * vb.x;  s[5] += vb.y; ss[5] += vb.y * vb.y;
      s[6] += vb.z; ss[6] += vb.z * vb.z;  s[7] += vb.w; ss[7] += vb.w * vb.w;
    }
    __builtin_amdgcn_s_wait_tensorcnt((short)0);   // half 1 landed
    asm volatile("" ::: "memory");
#pragma unroll 4
    for (int k = 0; k < 16; ++k) {
      const float4 va = *(const float4*)(gbuf + 4096 + k * EDIM + lane * 8);
      const float4 vb = *(const float4*)(gbuf + 4096 + k * EDIM + lane * 8 + 4);
      s[0] += va.x; ss[0] += va.x * va.x;  s[1] += va.y; ss[1] += va.y * va.y;
      s[2] += va.z; ss[2] += va.z * va.z;  s[3] += va.w; ss[3] += va.w * va.w;
      s[4] += vb.x; ss[4] += vb.x * vb.x;  s[5] += vb.y; ss[5] += vb.y * vb.y;
      s[6] += vb.z; ss[6] += vb.z * vb.z;  s[7] += vb.w; ss[7] += vb.w * vb.w;
    }

    float con = 0.f;
#pragma unroll
    for (int j = 0; j < 8; ++j) {
      const float m = s[j] * (1.0f / (float)NHN);
      float var = (ss[j] - (float)NHN * m * m) * (1.0f / (float)(NHN - 1));
      var = var > 0.f ? var : 0.f;
      con += sqrtf(var) * invg[j];
    }
#pragma unroll
    for (int off = 16; off > 0; off >>= 1) con += __shfl_xor(con, off);
    if (lane == 0) out_ld[b * NPTS + qbase + qq] = con;
  }
}

// Stage 4: full bitonic sort (descending) of 8192 keys per batch in LDS.
// key = (dist_bits << 32) | (8191 - i)  -> desc dist, ties: lower index first.
__global__ __launch_bounds__(1024) void topk_sort_kernel(
    const float* __restrict__ ld, int* __restrict__ keep) {
  extern __shared__ unsigned long long key[];
  const int b = blockIdx.x, tid = threadIdx.x;
  for (int i = tid; i < NPTS; i += 1024) {
    const unsigned int db = __float_as_uint(ld[b * NPTS + i]); // dist >= 0
    key[i] = ((unsigned long long)db << 32) | (unsigned int)(NPTS - 1 - i);
  }
  __syncthreads();
  for (int k = 2; k <= NPTS; k <<= 1) {
    for (int j = k >> 1; j > 0; j >>= 1) {
      for (int i = tid; i < NPTS; i += 1024) {
        const int ij = i ^ j;
        if (ij > i) {
          const unsigned long long a = key[i], c = key[ij];
          const bool descRegion = ((i & k) == 0);
          if (descRegion ? (a < c) : (a > c)) { key[i] = c; key[ij] = a; }
        }
      }
      __syncthreads();
    }
  }
  for (int i = tid; i < NKEEP; i += 1024)
    keep[b * NKEEP + i] = (NPTS - 1) - (int)(key[i] & 0xFFFFFFFFu);
}

// Stage 5: gather kept rows of x and coords into d_out.
__global__ void gather_out_kernel(const float* __restrict__ x,
                                  const float* __restrict__ coords,
                                  const int* __restrict__ keep,
                                  float* __restrict__ out) {
  const int b = blockIdx.y, tid = threadIdx.x;
  float* outx = out;                                      // [2,4096,256]
  float* outc = out + (size_t)BATCH * NKEEP * EDIM;       // [2,2,4096,1]
  const int r0 = blockIdx.x * 16;
  for (int rr = 0; rr < 16; ++rr) {
    const int row = r0 + rr;
    const int idx = keep[b * NKEEP + row];
    outx[(((size_t)b * NKEEP + row) << 8) + tid] =
        x[(((size_t)b * NPTS + idx) << 8) + tid];
  }
  if (tid < 32) {
    const int rr = tid & 15, cc = tid >> 4;
    const int row = r0 + rr;
    const int idx = keep[b * NKEEP + row];
    outc[(size_t)b * 2 * NKEEP + (size_t)cc * NKEEP + row] =
        coords[(size_t)b * 2 * NPTS + (size_t)cc * NPTS + idx];
  }
}

extern "C" void kernel_launch(void* const* d_in, const int* in_sizes, int n_in,
                              void* d_out, int out_size, void* d_ws, size_t ws_size,
                              hipStream_t stream) {
  const float* x      = (const float*)d_in[0];  // [2,8192,256]
  const float* coords = (const float*)d_in[1];  // [2,2,8192,1]
  float* out = (float*)d_out;
  float* ws  = (float*)d_ws;

  float* part  = ws;                // 8192 floats
  float* invgm = ws + 8192;         // 512 floats
  int*   keep  = (int*)(ws + 8704); // 8192 ints

  // local_dist lives in its final d_out slot: after x_out and coords_out
  float* out_ld = out + (size_t)BATCH * NKEEP * EDIM + (size_t)BATCH * 2 * NKEEP;

  colsum_part_kernel<<<dim3(16, BATCH), 256, 0, stream>>>(x, part);
  make_invgm_kernel<<<1, 512, 0, stream>>>(part, invgm);
  knn_std_kernel<<<BATCH * (NPTS / 64), 128, 4 * 8192 * sizeof(float), stream>>>(
      x, coords, invgm, out_ld);
  topk_sort_kernel<<<BATCH, 1024, (size_t)NPTS * sizeof(unsigned long long),
                     stream>>>(out_ld, keep);
  gather_out_kernel<<<dim3(NKEEP / 16, BATCH), 256, 0, stream>>>(x, coords, keep,
                                                                 out);
}